// CMR_59931973648949
// MI455X (gfx1250) — compile-verified
//
#include <hip/hip_runtime.h>
#include <math.h>

// Problem constants (match reference)
#define BB   16
#define KK   12
#define RR   12
#define NN   64
#define DW   512
#define DV   512
#define DVIS 2048
#define DE   512
#define DC   1024

typedef __attribute__((ext_vector_type(2))) float v2f;
typedef __attribute__((ext_vector_type(8))) float v8f;

// GCC-style vector to match the async-to-LDS builtin's parameter type exactly.
typedef int avec4i __attribute__((vector_size(16)));
#define AS1 __attribute__((address_space(1)))
#define AS3 __attribute__((address_space(3)))

#define INV_SQRT_512 0.04419417382415922f

// ---------------------------------------------------------------------------
// CDNA5 async copy: 16B per lane, global -> LDS, tracked by ASYNCcnt.
// ---------------------------------------------------------------------------
__device__ __forceinline__ void async_copy_b128(const float* g, float* l) {
#if __has_builtin(__builtin_amdgcn_global_load_async_to_lds_b128)
  __builtin_amdgcn_global_load_async_to_lds_b128(
      (AS1 avec4i*)g, (AS3 avec4i*)l, 0, 0);
#else
  unsigned long long ga = (unsigned long long)g;
  unsigned           lo = (unsigned)(unsigned long long)l;  // LDS byte offset
  asm volatile("global_load_async_to_lds_b128 %0, %1, off"
               :: "v"(lo), "v"(ga) : "memory");
#endif
}

__device__ __forceinline__ void wait_async0() {
#if __has_builtin(__builtin_amdgcn_s_wait_asynccnt)
  __builtin_amdgcn_s_wait_asynccnt(0);
#else
  asm volatile("s_wait_asynccnt 0" ::: "memory");
#endif
}

// ---------------------------------------------------------------------------
// C = A * W^T (+bias) on the WMMA pipe with LDS-staged weight tiles.
//   A: [M, Kdim] row-major, lda ; W: [Nout, Kdim] row-major, ldw
//   C: [M, Nout] row-major, ldc
// Block = nwaves waves. Each wave owns a 16-row M tile; all waves share one
// 64-column N tile whose weight chunk (64 cols x 64 K) is staged in LDS via
// async-to-LDS loads. grid = (Mtiles/nwaves, Nout/64).
// Requires: M%16==0, Nout%64==0, Kdim%64==0.
// ---------------------------------------------------------------------------
#define KC 64
#define BSTR 68   // padded LDS row stride (floats): bank = 4c+k mod 64 -> conflict-free

__global__ void gemm_f32_wmma(
    const float* __restrict__ A, int lda,
    const float* __restrict__ W, int ldw,
    float* __restrict__ C, int ldc,
    int Kdim, const float* __restrict__ bias)
{
  __shared__ float ldsB[KC * BSTR];

  const int tid    = threadIdx.x;
  const int lane   = tid & 31;
  const int wave   = tid >> 5;
  const int nwaves = blockDim.x >> 5;
  const int nthr   = blockDim.x;

  const int m0   = (blockIdx.x * nwaves + wave) * 16;
  const int n0   = blockIdx.y * 64;
  const int r    = lane & 15;
  const int koff = (lane >> 4) * 2;        // 0 or 2

  const float* Arow = A + (long)(m0 + r) * lda + koff;

  v8f acc0 = {}, acc1 = {}, acc2 = {}, acc3 = {};

  for (int k0 = 0; k0 < Kdim; k0 += KC) {
    // ---- cooperative async fill: W[n0+c][k0 .. k0+63] -> ldsB[c][0..63]
    for (int idx = tid; idx < KC * 16; idx += nthr) {
      const int c  = idx >> 4;             // 0..63 (column within tile)
      const int qd = (idx & 15) * 4;       // 16B quad within the 64-float row
      async_copy_b128(W + (long)(n0 + c) * ldw + k0 + qd,
                      &ldsB[c * BSTR + qd]);
    }
    wait_async0();
    __syncthreads();

    // ---- compute on the staged chunk
#pragma unroll 4
    for (int kk = 0; kk < KC; kk += 4) {
      v2f a = *(const v2f*)(Arow + k0 + kk);
      const float* bb = &ldsB[r * BSTR + kk + koff];
      v2f b0 = *(const v2f*)(bb);
      v2f b1 = *(const v2f*)(bb + 16 * BSTR);
      v2f b2 = *(const v2f*)(bb + 32 * BSTR);
      v2f b3 = *(const v2f*)(bb + 48 * BSTR);
      acc0 = __builtin_amdgcn_wmma_f32_16x16x4_f32(false, a, false, b0, (short)0, acc0, false, false);
      acc1 = __builtin_amdgcn_wmma_f32_16x16x4_f32(false, a, false, b1, (short)0, acc1, false, false);
      acc2 = __builtin_amdgcn_wmma_f32_16x16x4_f32(false, a, false, b2, (short)0, acc2, false, false);
      acc3 = __builtin_amdgcn_wmma_f32_16x16x4_f32(false, a, false, b3, (short)0, acc3, false, false);
    }
    __syncthreads();
  }

  // C/D layout: VGPR j -> row m0 + j + 8*(lane>>4); col n0 + t*16 + (lane&15)
  const int mrow = m0 + (lane >> 4) * 8;
  v8f accs[4] = {acc0, acc1, acc2, acc3};
#pragma unroll
  for (int t = 0; t < 4; ++t) {
    const int col = n0 + t * 16 + r;
    const float bv = bias ? bias[col] : 0.0f;
#pragma unroll
    for (int j = 0; j < 8; ++j) {
      C[(long)(mrow + j) * ldc + col] = accs[t][j] + bv;
    }
  }
}

// ---------------------------------------------------------------------------
// scale = mean(norm_w)   (512 elements, one block of 256)
// ---------------------------------------------------------------------------
__global__ __launch_bounds__(256) void k_scale(const float* __restrict__ norm_w,
                                               float* __restrict__ out)
{
  __shared__ float red[256];
  int t = threadIdx.x;
  red[t] = norm_w[t] + norm_w[t + 256];
  __syncthreads();
  for (int off = 128; off > 0; off >>= 1) {
    if (t < off) red[t] += red[t + off];
    __syncthreads();
  }
  if (t == 0) out[0] = red[0] * (1.0f / (float)DV);
}

// ---------------------------------------------------------------------------
// feat[b,n,:] = vision_feat[b,:,n] / max(||.||,1e-12) * scale
// One block per batch b (256 threads): coalesced lane access over n, and all
// touched cachelines fully consumed within the block.
// ---------------------------------------------------------------------------
__global__ __launch_bounds__(256) void k_normalize(const float* __restrict__ vf,
                                                   const float* __restrict__ scale,
                                                   float* __restrict__ feat)
{
  const int b = blockIdx.x;
  const int tid = threadIdx.x;
  const int n = tid & 63;
  const int part = tid >> 6;                       // 0..3
  const float* base = vf + (long)b * DVIS * NN;

  float ss = 0.0f;
  const int v0 = part * (DVIS / 4);
  for (int v = v0; v < v0 + DVIS / 4; ++v) {
    float x = base[(long)v * NN + n];
    ss += x * x;
  }
  __shared__ float red[4][64];
  __shared__ float invn[64];
  red[part][n] = ss;
  __syncthreads();
  if (tid < 64) {
    float t = red[0][tid] + red[1][tid] + red[2][tid] + red[3][tid];
    invn[tid] = scale[0] / fmaxf(sqrtf(t), 1e-12f);
  }
  __syncthreads();

  float* dst = feat + (long)b * NN * DVIS;
  for (int idx = tid; idx < NN * DVIS; idx += 256) {
    const int nn2 = idx >> 11;                     // / DVIS
    const int v = idx & (DVIS - 1);
    dst[idx] = base[(long)v * NN + nn2] * invn[nn2];
  }
}

// ---------------------------------------------------------------------------
// find[b,k,n] = node_mask[b,k] * softmax_n( q[b,k]·feat_v[b,n] / sqrt(DV), box_mask )
// grid = B*K blocks, 64 threads (thread = n)
// ---------------------------------------------------------------------------
__global__ __launch_bounds__(64) void k_find(const float* __restrict__ q,
                                             const float* __restrict__ feat_v,
                                             const float* __restrict__ box_mask,
                                             const float* __restrict__ node_mask,
                                             float* __restrict__ find)
{
  const int bk = blockIdx.x;
  const int b = bk / KK, k = bk % KK;
  const int n = threadIdx.x;

  const float* qr = q + (long)bk * DV;
  const float* fv = feat_v + ((long)b * NN + n) * DV;
  float s = 0.0f;
  for (int d = 0; d < DV; ++d) s += qr[d] * fv[d];
  s *= INV_SQRT_512;
  if (!(box_mask[b * NN + n] > 0.0f)) s = -1e9f;

  __shared__ float red[64];
  red[n] = s;
  __syncthreads();
  for (int off = 32; off > 0; off >>= 1) {
    if (n < off) red[n] = fmaxf(red[n], red[n + off]);
    __syncthreads();
  }
  const float mx = red[0];
  __syncthreads();
  const float e = __expf(s - mx);
  red[n] = e;
  __syncthreads();
  for (int off = 32; off > 0; off >>= 1) {
    if (n < off) red[n] += red[n + off];
    __syncthreads();
  }
  find[(long)bk * NN + n] = (e / red[0]) * node_mask[b * KK + k];
}

// ---------------------------------------------------------------------------
// s0[b,r,j] = p0[b,j]·qe[b,r] ; s1[b,r,j] = p1[b,j]·qe[b,r]
// grid = B*R blocks, 64 threads (thread = j)
// ---------------------------------------------------------------------------
__global__ __launch_bounds__(64) void k_s01(const float* __restrict__ p0,
                                            const float* __restrict__ p1,
                                            const float* __restrict__ qe,
                                            float* __restrict__ s0,
                                            float* __restrict__ s1)
{
  const int br = blockIdx.x;
  const int b = br / RR;
  const int j = threadIdx.x;
  const float* qer = qe + (long)br * DE;
  const float* p0r = p0 + ((long)b * NN + j) * DE;
  const float* p1r = p1 + ((long)b * NN + j) * DE;
  float a = 0.0f, c = 0.0f;
  for (int e = 0; e < DE; ++e) {
    const float w = qer[e];
    a += p0r[e] * w;
    c += p1r[e] * w;
  }
  s0[(long)br * NN + j] = a;
  s1[(long)br * NN + j] = c;
}

// ---------------------------------------------------------------------------
// find2[b,k,m] = find[b,k,m] + sum_{r: subj[b,r]==k} relate_mask[b,r] *
//                sum_n find[b,obj,n] * sigmoid((s0[b,r,m]+s1[b,r,n])/sqrt(DE))
//                             * relation_mask[b,n,m]
// Scatter rewritten as gather over r -> deterministic (no atomics).
// grid = B*K blocks, 64 threads (thread = m)
// ---------------------------------------------------------------------------
__global__ __launch_bounds__(64) void k_scatter(const float* __restrict__ find,
                                                const float* __restrict__ s0,
                                                const float* __restrict__ s1,
                                                const float* __restrict__ relation_mask,
                                                const float* __restrict__ relate_mask,
                                                const int* __restrict__ relate_os,
                                                float* __restrict__ find2)
{
  const int bk = blockIdx.x;
  const int b = bk / KK, k = bk % KK;
  const int m = threadIdx.x;

  float acc = find[(long)bk * NN + m];
  const float* rmcol = relation_mask + (long)b * NN * NN + m;  // [n, m], stride NN over n

  for (int r = 0; r < RR; ++r) {
    const int br = b * RR + r;
    const int subj = relate_os[br * 2 + 1];
    if (subj != k) continue;                 // subj==-1 never matches k>=0
    int obj = relate_os[br * 2 + 0];
    if (obj < 0) obj = 0;                    // jnp.clip(obj, 0)
    const float s0m = s0[(long)br * NN + m];
    const float* s1r = s1 + (long)br * NN;
    const float* fr = find + ((long)b * KK + obj) * NN;
    float c = 0.0f;
    for (int n = 0; n < NN; ++n) {
      const float lg = (s0m + s1r[n]) * INV_SQRT_512;
      const float att = (1.0f / (1.0f + __expf(-lg))) * rmcol[n * NN];
      c += fr[n] * att;
    }
    acc += c * relate_mask[br];
  }
  find2[(long)bk * NN + m] = acc;
}

// ---------------------------------------------------------------------------
// final_att[b,n] = maxk find2 / clamp(max_n, >=1), masked
// grid = B blocks, 64 threads
// ---------------------------------------------------------------------------
__global__ __launch_bounds__(64) void k_final(const float* __restrict__ find2,
                                              const float* __restrict__ box_mask,
                                              float* __restrict__ final_att)
{
  const int b = blockIdx.x;
  const int n = threadIdx.x;
  float v = find2[((long)b * KK) * NN + n];
  for (int k = 1; k < KK; ++k)
    v = fmaxf(v, find2[((long)b * KK + k) * NN + n]);

  __shared__ float red[64];
  red[n] = v;
  __syncthreads();
  for (int off = 32; off > 0; off >>= 1) {
    if (n < off) red[n] = fmaxf(red[n], red[n + off]);
    __syncthreads();
  }
  float norm = red[0];
  norm = (norm <= 1.0f) ? 1.0f : norm;
  const float bm = box_mask[b * NN + n];
  final_att[b * NN + n] = (v / norm) * bm + (1.0f - bm) * 1e-7f;
}

// ---------------------------------------------------------------------------
// mem[b,v] = sum_n final_att[b,n] * vision_feat[b,v,n]
// grid = (DVIS/256, B), 256 threads
// ---------------------------------------------------------------------------
__global__ __launch_bounds__(256) void k_mem(const float* __restrict__ final_att,
                                             const float* __restrict__ vf,
                                             float* __restrict__ mem)
{
  const int b = blockIdx.y;
  const int v = blockIdx.x * 256 + threadIdx.x;
  const float* row = vf + ((long)b * DVIS + v) * NN;
  const float* att = final_att + b * NN;
  float s = 0.0f;
  for (int n = 0; n < NN; ++n) s += att[n] * row[n];
  mem[(long)b * DVIS + v] = s;
}

// ---------------------------------------------------------------------------
extern "C" void kernel_launch(void* const* d_in, const int* in_sizes, int n_in,
                              void* d_out, int out_size, void* d_ws, size_t ws_size,
                              hipStream_t stream) {
  (void)in_sizes; (void)n_in; (void)out_size; (void)ws_size;

  const float* node_rep      = (const float*)d_in[0];   // [B,K,DW]
  const float* relate_rep    = (const float*)d_in[1];   // [B,R,DW]
  const int*   relate_os     = (const int*)d_in[2];     // [B,R,2] (obj,subj)
  const float* relate_mask   = (const float*)d_in[3];   // [B,R]
  const float* vision_feat   = (const float*)d_in[4];   // [B,DVIS,N]
  const float* relation_mask = (const float*)d_in[5];   // [B,N,N]
  const float* box_mask      = (const float*)d_in[6];   // [B,N]
  const float* node_mask     = (const float*)d_in[7];   // [B,K]
  const float* norm_w        = (const float*)d_in[8];   // [DV]
  const float* W_v           = (const float*)d_in[9];   // [DV,DVIS]
  const float* W_e           = (const float*)d_in[10];  // [DE,2*DV]
  const float* W_node        = (const float*)d_in[11];  // [DV,DW]
  const float* W_rel         = (const float*)d_in[12];  // [DE,DW]
  const float* W_out         = (const float*)d_in[13];  // [DC,DVIS]
  const float* b_out         = (const float*)d_in[14];  // [DC]
  float* out = (float*)d_out;                           // [B,DC]

  // Workspace layout (floats)
  float* ws = (float*)d_ws;
  float* w_scale = ws;                         // 16
  float* w_feat  = w_scale + 16;               // B*N*DVIS  = 2097152
  float* w_featv = w_feat  + (long)BB*NN*DVIS; // B*N*DV    = 524288
  float* w_p0    = w_featv + (long)BB*NN*DV;   // B*N*DE
  float* w_p1    = w_p0    + (long)BB*NN*DE;   // B*N*DE
  float* w_q     = w_p1    + (long)BB*NN*DE;   // B*K*DV
  float* w_qe    = w_q     + (long)BB*KK*DV;   // B*R*DE
  float* w_find  = w_qe    + (long)BB*RR*DE;   // B*K*N
  float* w_find2 = w_find  + (long)BB*KK*NN;   // B*K*N
  float* w_s0    = w_find2 + (long)BB*KK*NN;   // B*R*N
  float* w_s1    = w_s0    + (long)BB*RR*NN;   // B*R*N
  float* w_att   = w_s1    + (long)BB*RR*NN;   // B*N
  float* w_mem   = w_att   + (long)BB*NN;      // B*DVIS

  // 1) scalar scale = mean(norm_w)
  k_scale<<<1, 256, 0, stream>>>(norm_w, w_scale);

  // 2) normalized + scaled features, transposed to [B,N,DVIS]
  k_normalize<<<BB, 256, 0, stream>>>(vision_feat, w_scale, w_feat);

  // 3) feat_v = feat · W_v^T     [1024,2048]x[512,2048]^T -> [1024,512]
  gemm_f32_wmma<<<dim3(8, DV / 64), 256, 0, stream>>>(
      w_feat, DVIS, W_v, DVIS, w_featv, DV, DVIS, nullptr);

  // 4) p0 = feat_v · W_e[:, :DV]^T ; p1 = feat_v · W_e[:, DV:]^T
  gemm_f32_wmma<<<dim3(8, DE / 64), 256, 0, stream>>>(
      w_featv, DV, W_e, 2 * DV, w_p0, DE, DV, nullptr);
  gemm_f32_wmma<<<dim3(8, DE / 64), 256, 0, stream>>>(
      w_featv, DV, W_e + DV, 2 * DV, w_p1, DE, DV, nullptr);

  // 5) q = node_rep · W_node^T ; qe = relate_rep · W_rel^T   [192,512]
  gemm_f32_wmma<<<dim3(3, DV / 64), 128, 0, stream>>>(
      node_rep, DW, W_node, DW, w_q, DV, DW, nullptr);
  gemm_f32_wmma<<<dim3(3, DE / 64), 128, 0, stream>>>(
      relate_rep, DW, W_rel, DW, w_qe, DE, DW, nullptr);

  // 6) masked softmax attention over boxes
  k_find<<<BB * KK, 64, 0, stream>>>(w_q, w_featv, box_mask, node_mask, w_find);

  // 7) factorized edge logits
  k_s01<<<BB * RR, 64, 0, stream>>>(w_p0, w_p1, w_qe, w_s0, w_s1);

  // 8) transfer + scatter (as deterministic gather)
  k_scatter<<<BB * KK, 64, 0, stream>>>(w_find, w_s0, w_s1, relation_mask,
                                        relate_mask, relate_os, w_find2);

  // 9) final attention
  k_final<<<BB, 64, 0, stream>>>(w_find2, box_mask, w_att);

  // 10) attention-weighted readout
  k_mem<<<dim3(DVIS / 256, BB), 256, 0, stream>>>(w_att, vision_feat, w_mem);

  // 11) final_mem = mem · W_out^T + b_out   [16,2048]x[1024,2048]^T -> [16,1024]
  gemm_f32_wmma<<<dim3(1, DC / 64), 32, 0, stream>>>(
      w_mem, DVIS, W_out, DVIS, out, DC, DVIS, b_out);
}